// GCN_71528385347624
// MI455X (gfx1250) — compile-verified
//
#include <hip/hip_runtime.h>
#include <hip/hip_bf16.h>

// ---------------------------------------------------------------------------
// GCN forward for MI455X (gfx1250).
// Dense GEMMs use V_WMMA_F32_16X16X4_F32 (wave32 matrix pipe, fp32-exact).
// SpMM (COO segment-sum) uses L2-resident float atomics: the whole working
// set (~100MB) fits in the 192MB L2, so gather/scatter traffic stays on-chip.
// ---------------------------------------------------------------------------

typedef __attribute__((ext_vector_type(2))) float v2f;
typedef __attribute__((ext_vector_type(8))) float v8f;

#define D_IN 128
#define D_H  64
#define D_OUT 40

// ---------------------------------------------------------------------------
// GEMM1: support[N,64] = x[N,128] @ W1[128,64]
// One wave per 16x16 output tile. blockDim = 128 (4 waves = 4 col tiles).
// WMMA f32 16x16x4 operand layout (ISA 7.12.2):
//   A (16x4,MxK):  lanes 0-15 -> M=lane, {K=0,K=1}; lanes 16-31 -> M=lane-16, {K=2,K=3}
//   B (4x16,KxN):  lanes 0-15 -> N=lane, {K=0,K=1}; lanes 16-31 -> N=lane-16, {K=2,K=3}
//   C/D (16x16):   vgpr v -> M=v (+8 for upper half-wave), N=lane&15
// ---------------------------------------------------------------------------
__global__ void gcn_gemm1(const float* __restrict__ x,
                          const float* __restrict__ W1,
                          float* __restrict__ support, int N) {
    const int wave = threadIdx.x >> 5;
    const int lane = threadIdx.x & 31;
    const int half = lane >> 4;        // 0: lanes 0-15, 1: lanes 16-31
    const int l    = lane & 15;

    const int row0 = blockIdx.x * 16;
    const int col0 = wave * 16;        // 4 waves cover 64 cols

    int rA = row0 + l;                 // row this lane loads for A
    if (rA >= N) rA = N - 1;           // clamp (N is a multiple of 16 anyway)

    v8f acc = {0.f, 0.f, 0.f, 0.f, 0.f, 0.f, 0.f, 0.f};

    const float* xrow = x + (size_t)rA * D_IN;

    for (int k = 0; k < D_IN; k += 4) {
        const int kk = k + 2 * half;
        // A: two consecutive K values for this lane's row (8B-aligned b64 load)
        v2f a = *(const v2f*)(xrow + kk);
        // B: W1 rows kk, kk+1 at column col0+l
        v2f b;
        b.x = W1[(size_t)kk       * D_H + col0 + l];
        b.y = W1[(size_t)(kk + 1) * D_H + col0 + l];
        acc = __builtin_amdgcn_wmma_f32_16x16x4_f32(
            /*neg_a=*/false, a, /*neg_b=*/false, b,
            /*c_mod=*/(short)0, acc, /*reuse_a=*/false, /*reuse_b=*/false);
    }

    #pragma unroll
    for (int v = 0; v < 8; ++v) {
        const int r = row0 + v + 8 * half;
        if (r < N) support[(size_t)r * D_H + col0 + l] = acc[v];
    }
}

// ---------------------------------------------------------------------------
// GEMM2: support2[N,40] = relu(hidden[N,64]) @ W2[64,40]   (ReLU fused in A load)
// blockDim = 96 (3 waves = 3 col tiles covering cols 0..47; cols >= 40 masked)
// ---------------------------------------------------------------------------
__global__ void gcn_gemm2(const float* __restrict__ hidden,
                          const float* __restrict__ W2,
                          float* __restrict__ support2, int N) {
    const int wave = threadIdx.x >> 5;
    const int lane = threadIdx.x & 31;
    const int half = lane >> 4;
    const int l    = lane & 15;

    const int row0 = blockIdx.x * 16;
    const int col0 = wave * 16;
    const int col  = col0 + l;
    const bool colOk = (col < D_OUT);

    int rA = row0 + l;
    if (rA >= N) rA = N - 1;

    v8f acc = {0.f, 0.f, 0.f, 0.f, 0.f, 0.f, 0.f, 0.f};
    const float* hrow = hidden + (size_t)rA * D_H;

    #pragma unroll 4
    for (int k = 0; k < D_H; k += 4) {
        const int kk = k + 2 * half;
        v2f a = *(const v2f*)(hrow + kk);
        a.x = fmaxf(a.x, 0.f);         // fused ReLU
        a.y = fmaxf(a.y, 0.f);
        v2f b;
        b.x = colOk ? W2[(size_t)kk       * D_OUT + col] : 0.f;
        b.y = colOk ? W2[(size_t)(kk + 1) * D_OUT + col] : 0.f;
        acc = __builtin_amdgcn_wmma_f32_16x16x4_f32(
            false, a, false, b, (short)0, acc, false, false);
    }

    if (colOk) {
        #pragma unroll
        for (int v = 0; v < 8; ++v) {
            const int r = row0 + v + 8 * half;
            if (r < N) support2[(size_t)r * D_OUT + col] = acc[v];
        }
    }
}

// ---------------------------------------------------------------------------
// Bias broadcast init: dst[n,d] = bias[d]
// ---------------------------------------------------------------------------
__global__ void gcn_bias_init(float* __restrict__ dst,
                              const float* __restrict__ bias,
                              int total, int d) {
    int i = blockIdx.x * blockDim.x + threadIdx.x;
    if (i < total) dst[i] = bias[i % d];
}

// ---------------------------------------------------------------------------
// SpMM scatter, 64 features: 16 threads/edge, float4 per thread.
// hidden[row[e], d..d+3] += vals[e] * support[col[e], d..d+3]
// ---------------------------------------------------------------------------
__global__ void gcn_scatter64(const int* __restrict__ erow,
                              const int* __restrict__ ecol,
                              const float* __restrict__ vals,
                              const float* __restrict__ support,
                              float* __restrict__ hidden, int E) {
    const int tid = blockIdx.x * blockDim.x + threadIdx.x;
    const int e = tid >> 4;
    if (e >= E) return;
    const int d = (tid & 15) * 4;
    const int r = erow[e];
    const int c = ecol[e];
    const float v = vals[e];
    const float4 s = *(const float4*)(support + (size_t)c * D_H + d);
    float* h = hidden + (size_t)r * D_H + d;
    atomicAdd(h + 0, v * s.x);
    atomicAdd(h + 1, v * s.y);
    atomicAdd(h + 2, v * s.z);
    atomicAdd(h + 3, v * s.w);
}

// ---------------------------------------------------------------------------
// SpMM scatter, 40 features: 10 threads/edge, float4 per thread.
// (stride 40 floats = 160B, +16B*q -> every float4 access is 16B aligned)
// ---------------------------------------------------------------------------
__global__ void gcn_scatter40(const int* __restrict__ erow,
                              const int* __restrict__ ecol,
                              const float* __restrict__ vals,
                              const float* __restrict__ support2,
                              float* __restrict__ out, int E) {
    const int tid = blockIdx.x * blockDim.x + threadIdx.x;
    const int e = tid / 10;
    if (e >= E) return;
    const int d = (tid - e * 10) * 4;
    const int r = erow[e];
    const int c = ecol[e];
    const float v = vals[e];
    const float4 s = *(const float4*)(support2 + (size_t)c * D_OUT + d);
    float* o = out + (size_t)r * D_OUT + d;
    atomicAdd(o + 0, v * s.x);
    atomicAdd(o + 1, v * s.y);
    atomicAdd(o + 2, v * s.z);
    atomicAdd(o + 3, v * s.w);
}

// ---------------------------------------------------------------------------
// Launch
// ---------------------------------------------------------------------------
extern "C" void kernel_launch(void* const* d_in, const int* in_sizes, int n_in,
                              void* d_out, int out_size, void* d_ws, size_t ws_size,
                              hipStream_t stream) {
    const float* x    = (const float*)d_in[0];
    const int*   erow = (const int*)  d_in[1];
    const int*   ecol = (const int*)  d_in[2];
    const float* vals = (const float*)d_in[3];
    const float* W1   = (const float*)d_in[4];
    const float* b1   = (const float*)d_in[5];
    const float* W2   = (const float*)d_in[6];
    const float* b2   = (const float*)d_in[7];
    float* out = (float*)d_out;

    const int N = in_sizes[0] / D_IN;   // 100000
    const int E = in_sizes[1];          // 1600000

    // Workspace layout (~67 MB): support | hidden | support2
    float* support  = (float*)d_ws;
    float* hidden   = support + (size_t)N * D_H;
    float* support2 = hidden  + (size_t)N * D_H;

    const int rowTiles = (N + 15) / 16;

    // 1) support = x @ W1           (WMMA f32)
    gcn_gemm1<<<rowTiles, 128, 0, stream>>>(x, W1, support, N);

    // 2) hidden = broadcast(b1)
    {
        const int total = N * D_H;
        gcn_bias_init<<<(total + 255) / 256, 256, 0, stream>>>(hidden, b1, total, D_H);
    }

    // 3) hidden += adj @ support    (L2 atomics)
    {
        const long long threads = (long long)E * 16;
        gcn_scatter64<<<(unsigned)((threads + 255) / 256), 256, 0, stream>>>(
            erow, ecol, vals, support, hidden, E);
    }

    // 4) support2 = relu(hidden) @ W2  (WMMA f32, ReLU fused)
    gcn_gemm2<<<rowTiles, 96, 0, stream>>>(hidden, W2, support2, N);

    // 5) out = broadcast(b2)
    {
        const int total = N * D_OUT;
        gcn_bias_init<<<(total + 255) / 256, 256, 0, stream>>>(out, b2, total, D_OUT);
    }

    // 6) out += adj @ support2      (L2 atomics)
    {
        const long long threads = (long long)E * 10;
        gcn_scatter40<<<(unsigned)((threads + 255) / 256), 256, 0, stream>>>(
            erow, ecol, vals, support2, out, E);
    }
}